// FeatureGraphDiffusionMLP_22728966930821
// MI455X (gfx1250) — compile-verified
//
#include <hip/hip_runtime.h>
#include <stdint.h>

// Sizes fixed by the reference
#define NB   128   // nodes per graph (N)
#define FDIM 64    // FD
#define ODIM 128   // OD
#define NH   4     // heads
#define HDIM 32    // OD/H

#define CHUNK_ROWS 64          // adjacency rows staged per pass
#define ROWPITCH_I 132         // ints per LDS row (528 B = 33*16B: aligned for b128, low-conflict)

typedef float v2f __attribute__((ext_vector_type(2)));
typedef float v8f __attribute__((ext_vector_type(8)));
typedef int   vi4 __attribute__((vector_size(16)));   // matches builtin's diagnosed pointee type

// Workspace layout (floats):
//  [0..15]             : head scalars ws[which*4+h], which: 0=A,1=B,2=C,3=D (pre-scaled)
//  [16..16+512)        : WvA[p][h] (p-major)
//  [16+512..16+512+128): constp[p]
#define WS_WV   16
#define WS_CP   (16 + ODIM*NH)

#if defined(__has_builtin)
#if __has_builtin(__builtin_amdgcn_global_load_async_to_lds_b128)
#define HAVE_ASYNC_LDS 1
#endif
#if __has_builtin(__builtin_amdgcn_s_wait_asynccnt)
#define HAVE_WAIT_ASYNC 1
#endif
#endif

// AS1 = global (printed as __device__ by clang/HIP), AS3 = LDS
#define AS1P(p) ((__attribute__((address_space(1))) vi4*)(uintptr_t)(p))
#define AS3P(p) ((__attribute__((address_space(3))) vi4*)(uintptr_t)(p))

__device__ __forceinline__ void wait_asynccnt0() {
#if defined(HAVE_WAIT_ASYNC)
  __builtin_amdgcn_s_wait_asynccnt(0);
#elif defined(HAVE_ASYNC_LDS)
  asm volatile("s_wait_asynccnt 0x0" ::: "memory");
#endif
}

// ---------------------------------------------------------------------------
// Kernel 0: fold all weights into the rank-1-collapsed coefficients.
// ---------------------------------------------------------------------------
__global__ void fgd_precompute(
    const float* __restrict__ w_fe, const float* __restrict__ b_fe,
    const float* __restrict__ wq,   const float* __restrict__ bq,
    const float* __restrict__ wk,   const float* __restrict__ bk,
    const float* __restrict__ wv,   const float* __restrict__ bv,
    const float* __restrict__ wo,   const float* __restrict__ bo,
    const float* __restrict__ temperature,
    float* __restrict__ ws)
{
  __shared__ float aq[ODIM], cq[ODIM], ak[ODIM], ck[ODIM], av[ODIM], cv[ODIM];
  const int o = threadIdx.x;  // 128 threads
  {
    float qa = 0.f, qc = 0.f, ka = 0.f, kc = 0.f, va = 0.f, vc = 0.f;
    for (int f = 0; f < FDIM; ++f) {
      const float wf = w_fe[f], bf = b_fe[f];
      const float q = wq[o*FDIM + f], k = wk[o*FDIM + f], v = wv[o*FDIM + f];
      qa = fmaf(q, wf, qa); qc = fmaf(q, bf, qc);
      ka = fmaf(k, wf, ka); kc = fmaf(k, bf, kc);
      va = fmaf(v, wf, va); vc = fmaf(v, bf, vc);
    }
    aq[o] = qa; cq[o] = qc + bq[o];
    ak[o] = ka; ck[o] = kc + bk[o];
    av[o] = va; cv[o] = vc + bv[o];
  }
  __syncthreads();

  const float scale = temperature[0] * rsqrtf((float)HDIM);
  if (o < 16) {
    const int h = o & 3, which = o >> 2, base = h * HDIM;
    const float* u = (which < 2) ? aq : cq;   // 0:aq*ak 1:aq*ck 2:cq*ak 3:cq*ck
    const float* w = (which & 1) ? ck : ak;
    float s = 0.f;
    for (int d = 0; d < HDIM; ++d) s = fmaf(u[base + d], w[base + d], s);
    ws[which*4 + h] = s * scale;
  }
  {
    const int p = o;
    for (int h = 0; h < NH; ++h) {
      float s = 0.f;
      for (int d = 0; d < HDIM; ++d)
        s = fmaf(wo[p*ODIM + h*HDIM + d], av[h*HDIM + d], s);
      ws[WS_WV + p*NH + h] = s;
    }
    float s = 0.f;
    for (int q = 0; q < ODIM; ++q) s = fmaf(wo[p*ODIM + q], cv[q], s);
    ws[WS_CP + p] = s + bo[p];
  }
}

// ---------------------------------------------------------------------------
// Kernel 1: one block per batch. Async-stage adjacency rows (2 chunks of 64)
// -> masked softmax scalars s[h][n] -> WMMA f32 16x16x4 (K==H==4) output
// projection -> in-register LayerNorm.
// ---------------------------------------------------------------------------
__global__ __launch_bounds__(256)
void fgd_main(const float* __restrict__ features,
              const int*   __restrict__ adjacency,
              const float* __restrict__ gamma,
              const float* __restrict__ beta,
              const float* __restrict__ ws,
              float* __restrict__ out)
{
  __shared__ int   s_adj[CHUNK_ROWS * ROWPITCH_I];  // 33.8 KB raw int32 rows
  __shared__ float s_x[NB];
  __shared__ float s_s[NH * NB];                    // s[h][n]

  const int b   = blockIdx.x;
  const int tid = threadIdx.x;

  if (tid < NB) s_x[tid] = features[(size_t)b*NB + tid];

  // head/row mapping for phase 1: 256 threads = 64 rows x 4 heads per chunk
  const int nloc = tid & (CHUNK_ROWS - 1);
  const int h    = tid >> 6;                 // uniform per wave
  const float Ah = ws[0 + h], Bh = ws[4 + h], Ch = ws[8 + h], Dh = ws[12 + h];

  for (int c = 0; c < NB / CHUNK_ROWS; ++c) {
    // ---- stage chunk c: adjacency rows [c*64, c*64+64) as raw int32 ----
    const int* gsrc = adjacency + (size_t)b * NB * NB + (size_t)c * CHUNK_ROWS * NB;
    // 2048 x 16B transfers; row r gets 32 chunks at LDS pitch 528B
    for (int i = tid; i < (CHUNK_ROWS * NB) / 4; i += 256) {
      const int r = i >> 5, q16 = i & 31;
#if defined(HAVE_ASYNC_LDS)
      __builtin_amdgcn_global_load_async_to_lds_b128(
          AS1P(gsrc + r*NB + q16*4),
          AS3P(&s_adj[r*ROWPITCH_I + q16*4]),
          /*offset=*/0, /*cpol=*/0);
#else
      const int4 v = *(const int4*)(gsrc + r*NB + q16*4);
      *(int4*)&s_adj[r*ROWPITCH_I + q16*4] = v;
#endif
    }
    wait_asynccnt0();
    __syncthreads();

    // ---- phase 1: single-pass masked softmax (scores clipped to [-10,10]) ----
    {
      const int n = c * CHUNK_ROWS + nloc;
      const float xn = s_x[n];
      const int* arow = &s_adj[nloc * ROWPITCH_I];
      float num = 0.f, den = 0.f, xsum = 0.f;
      for (int m = 0; m < NB; ++m) {
        const float xm = s_x[m];
        xsum += xm;
        if (arow[m]) {
          float sc = fmaf(xn * xm, Ah, fmaf(xn, Bh, fmaf(xm, Ch, Dh)));
          sc = fminf(10.f, fmaxf(-10.f, sc));
          const float w = __expf(sc);
          den += w;
          num = fmaf(w, xm, num);
        }
      }
      // all-masked row -> reference softmax is uniform -> s = mean(x)
      s_s[h*NB + n] = den > 0.f ? num / den : xsum * (1.0f / NB);
    }
    __syncthreads();   // protect s_adj reuse in next chunk
  }

  // ---- phase 2: out[n,p] = sum_h s[h,n]*WvA[p,h] + const[p] via WMMA, then LN ----
  {
    const int wave = tid >> 5, lane = tid & 31;
    const int hh = lane >> 4, l = lane & 15;   // lane-half, lane-in-half
    const int n0 = wave * 16;                  // 8 waves -> 8 n-tiles

    // A tile 16x4 f32: lanes 0-15 hold K=0(V0),1(V1); lanes 16-31 hold K=2,3
    v2f a;
    a.x = s_s[(hh*2 + 0)*NB + n0 + l];
    a.y = s_s[(hh*2 + 1)*NB + n0 + l];

    v8f d[8];
#pragma unroll
    for (int t = 0; t < 8; ++t) {
      // B tile 4x16: lanes 0-15: K=0(V0),1(V1); lanes 16-31: K=2,3; N = lane%16
      const float* wvp = ws + WS_WV + (t*16 + l) * NH;
      v2f bm;
      bm.x = wvp[hh*2 + 0];
      bm.y = wvp[hh*2 + 1];
      v8f cz = {0.f,0.f,0.f,0.f,0.f,0.f,0.f,0.f};
      d[t] = __builtin_amdgcn_wmma_f32_16x16x4_f32(
          /*neg_a=*/false, a, /*neg_b=*/false, bm,
          /*c_mod=*/(short)0, cz, /*reuse_a=*/false, /*reuse_b=*/false);
      const float cp = ws[WS_CP + t*16 + l];
#pragma unroll
      for (int r = 0; r < 8; ++r) d[t][r] += cp;
    }

    // LayerNorm over p (row n lives in 16 lanes x 8 tiles for VGPR row r)
    float* outb = out + (size_t)b * NB * ODIM;
#pragma unroll
    for (int r = 0; r < 8; ++r) {
      float s1 = 0.f, s2 = 0.f;
#pragma unroll
      for (int t = 0; t < 8; ++t) { const float v = d[t][r]; s1 += v; s2 = fmaf(v, v, s2); }
#pragma unroll
      for (int off = 1; off < 16; off <<= 1) {   // reduce within each 16-lane half
        s1 += __shfl_xor(s1, off, 32);
        s2 += __shfl_xor(s2, off, 32);
      }
      const float mean = s1 * (1.0f / ODIM);
      const float var  = s2 * (1.0f / ODIM) - mean * mean;
      const float rstd = rsqrtf(var + 1e-5f);
      const int row = n0 + r + hh*8;
#pragma unroll
      for (int t = 0; t < 8; ++t) {
        const int col = t*16 + l;
        outb[row*ODIM + col] =
            fmaf((d[t][r] - mean) * rstd, gamma[col], beta[col]);
      }
    }
  }
}

// ---------------------------------------------------------------------------
extern "C" void kernel_launch(void* const* d_in, const int* in_sizes, int n_in,
                              void* d_out, int out_size, void* d_ws, size_t ws_size,
                              hipStream_t stream) {
  (void)n_in; (void)out_size; (void)ws_size;
  const float* features    = (const float*)d_in[0];
  const int*   adjacency   = (const int*)  d_in[1];
  const float* w_fe        = (const float*)d_in[2];
  const float* b_fe        = (const float*)d_in[3];
  const float* wq          = (const float*)d_in[4];
  const float* bq          = (const float*)d_in[5];
  const float* wk          = (const float*)d_in[6];
  const float* bk          = (const float*)d_in[7];
  const float* wv          = (const float*)d_in[8];
  const float* bv          = (const float*)d_in[9];
  const float* wo          = (const float*)d_in[10];
  const float* bo          = (const float*)d_in[11];
  const float* gamma       = (const float*)d_in[12];
  const float* beta        = (const float*)d_in[13];
  const float* temperature = (const float*)d_in[14];
  float* ws  = (float*)d_ws;
  float* op  = (float*)d_out;

  const int B = in_sizes[0] / NB;

  hipLaunchKernelGGL(fgd_precompute, dim3(1), dim3(128), 0, stream,
                     w_fe, b_fe, wq, bq, wk, bk, wv, bv, wo, bo, temperature, ws);
  hipLaunchKernelGGL(fgd_main, dim3(B), dim3(256), 0, stream,
                     features, adjacency, gamma, beta, ws, op);
}